// UpsampleNd_30846455119938
// MI455X (gfx1250) — compile-verified
//
#include <hip/hip_runtime.h>

// Nearest-neighbor 2x upsample: out[b,c,oh,ow] = x[b,c,oh/2,ow/2]
// x: [8,256,128,128] f32  ->  out: [8,256,256,256] f32
// Pure bandwidth kernel: ~671 MB HBM traffic, floor ~29us at 23.3 TB/s.

typedef float v4f __attribute__((ext_vector_type(4)));
typedef int   v4i __attribute__((ext_vector_type(4)));

#define AS1 __attribute__((address_space(1)))
#define AS3 __attribute__((address_space(3)))

typedef AS1 v4i* gvec_ptr;   // global (addrspace 1) 128-bit pointer
typedef AS3 v4i* lvec_ptr;   // LDS    (addrspace 3) 128-bit pointer

constexpr int kB = 8, kC = 256, kH = 128, kW = 128;
constexpr int kW4 = kW / 4;                                   // 32 float4 per input row
constexpr long long kM4 = (long long)kB * kC * kH * kW4;      // 8,388,608 input float4s
constexpr int BLOCK = 256;                                    // 8 waves (wave32)
constexpr int GRID  = 1024;
constexpr int ITERS = (int)(kM4 / ((long long)BLOCK * GRID)); // 32, exact
static_assert((long long)BLOCK * GRID * ITERS == kM4, "exact tiling, no tail");

__global__ __launch_bounds__(BLOCK) void upsample2x_nn_kernel(
    const float* __restrict__ x, float* __restrict__ y) {
  __shared__ v4f buf[2][BLOCK];   // 8 KiB double buffer, one float4 per thread per stage

  const int tid = threadIdx.x;
  const v4f* __restrict__ in4  = (const v4f*)x;
  v4f* __restrict__       out4 = (v4f*)y;

  const long long i0     = (long long)blockIdx.x * BLOCK + tid;
  const long long stride = (long long)GRID * BLOCK;

#if defined(__gfx1250__) && __has_builtin(__builtin_amdgcn_global_load_async_to_lds_b128) && \
    __has_builtin(__builtin_amdgcn_s_wait_asynccnt)
  // ---- CDNA5 async global->LDS pipeline (ASYNCcnt-tracked data mover) ----
  // Prologue: stage 0 in flight.
  __builtin_amdgcn_global_load_async_to_lds_b128(
      (gvec_ptr)(in4 + i0), (lvec_ptr)&buf[0][tid], 0, 0);

  for (int t = 0; t < ITERS; ++t) {
    const long long i = i0 + (long long)t * stride;
    if (t + 1 < ITERS) {
      // Issue next stage, then wait until only it remains outstanding.
      __builtin_amdgcn_global_load_async_to_lds_b128(
          (gvec_ptr)(in4 + i + stride), (lvec_ptr)&buf[(t + 1) & 1][tid], 0, 0);
      __builtin_amdgcn_s_wait_asynccnt(1);
    } else {
      __builtin_amdgcn_s_wait_asynccnt(0);
    }

    // Each lane reads back exactly the float4 its own async op deposited:
    // wave-local dependency, covered by s_wait_asynccnt -> no barrier needed.
    v4f v = buf[t & 1][tid];                                  // ds_load_b128

    const int ii   = (int)i;          // < 2^23, fits
    const int row  = ii >> 5;         // input row index   (i / 32)
    const int col4 = ii & 31;         // float4 col in row
    const int p    = row >> 7;        // (b,c) plane       (row / 128)
    const int h    = row & 127;       // input row in plane

    // Output float4 offsets: plane stride 256*256/4, row 2h -> 2h*256/4 = 128h,
    // col 2*(4*col4) floats -> 2*col4 float4s; next output row at +64.
    const long long o = (long long)p * 16384 + (long long)h * 128 + col4 * 2;

    v4f lo = {v.x, v.x, v.y, v.y};
    v4f hi = {v.z, v.z, v.w, v.w};
    // Streaming 512 MiB >> 192 MiB L2: non-temporal stores.
    __builtin_nontemporal_store(lo, out4 + o);
    __builtin_nontemporal_store(hi, out4 + o + 1);
    __builtin_nontemporal_store(lo, out4 + o + 64);
    __builtin_nontemporal_store(hi, out4 + o + 65);
  }
#else
  // ---- Fallback: direct b128 load path ----
  for (int t = 0; t < ITERS; ++t) {
    const long long i = i0 + (long long)t * stride;
    v4f v = __builtin_nontemporal_load(in4 + i);

    const int ii   = (int)i;
    const int row  = ii >> 5;
    const int col4 = ii & 31;
    const int p    = row >> 7;
    const int h    = row & 127;
    const long long o = (long long)p * 16384 + (long long)h * 128 + col4 * 2;

    v4f lo = {v.x, v.x, v.y, v.y};
    v4f hi = {v.z, v.z, v.w, v.w};
    __builtin_nontemporal_store(lo, out4 + o);
    __builtin_nontemporal_store(hi, out4 + o + 1);
    __builtin_nontemporal_store(lo, out4 + o + 64);
    __builtin_nontemporal_store(hi, out4 + o + 65);
  }
#endif
}

extern "C" void kernel_launch(void* const* d_in, const int* in_sizes, int n_in,
                              void* d_out, int out_size, void* d_ws, size_t ws_size,
                              hipStream_t stream) {
  (void)in_sizes; (void)n_in; (void)out_size; (void)d_ws; (void)ws_size;
  const float* x = (const float*)d_in[0];   // [8,256,128,128] f32
  // d_in[1] (kernel2d) is mathematically the identity-upsample stencil; unused.
  float* y = (float*)d_out;                 // [8,256,256,256] f32
  upsample2x_nn_kernel<<<GRID, BLOCK, 0, stream>>>(x, y);
}